// KnowledgeGraph_39822936769253
// MI455X (gfx1250) — compile-verified
//
#include <hip/hip_runtime.h>
#include <hip/hip_bf16.h>
#include <float.h>

// ---------------------------------------------------------------------------
// KG scoring: scores[b,e] = relu(ph[b,:] + (ent_emb @ W1t)[e,:]) . W2 + b2
// then top-k per batch.  D=128, H=256 fixed by the model; B,E,k from sizes.
//
// Fused kernel: M=64 entities per block (amortizes W1t L2 traffic 4x vs M=16),
// 4 waves, each wave owns 64 N columns; WMMA f32 16x16x4 (exact fp32 path).
// ---------------------------------------------------------------------------

#define DD   128          // embedding dim
#define HH   256          // hidden dim
#define MT   64           // entities per block
#define AS_STRIDE 132     // A tile pad: LDS bank = (4m+k)%64, conflict-free
#define PT_STRIDE 260     // 16x256 pt subtile pad
#define PH_STRIDE 260     // 8x256 ph tile pad

typedef __attribute__((ext_vector_type(2))) float v2f;
typedef __attribute__((ext_vector_type(8))) float v8f;

// ---------------------------------------------------------------------------
// Kernel 1: ph[b,j] = sum_d h[b,d]*W1[d,j] + r[b,d]*W1[128+d,j] + b1[j]
// ---------------------------------------------------------------------------
__global__ __launch_bounds__(256) void kg_ph_kernel(
    const int* __restrict__ head, const int* __restrict__ relation,
    const float* __restrict__ ent, const float* __restrict__ rel,
    const float* __restrict__ W1, const float* __restrict__ b1,
    float* __restrict__ ph)
{
    const int b = blockIdx.x;
    const int j = threadIdx.x;
    const float* hv = ent + (size_t)head[b] * DD;       // scalar-broadcast loads
    const float* rv = rel + (size_t)relation[b] * DD;
    float acc = b1[j];
    #pragma unroll 4
    for (int d = 0; d < DD; ++d) {
        acc = fmaf(hv[d], W1[(size_t)d * HH + j], acc);          // W1h
        acc = fmaf(rv[d], W1[(size_t)(DD + d) * HH + j], acc);   // W1r
    }
    ph[(size_t)b * HH + j] = acc;
}

// ---------------------------------------------------------------------------
// Kernel 2: fused pt-GEMM (WMMA) + relu-dot scoring for a 64-entity tile.
// ---------------------------------------------------------------------------
__global__ __launch_bounds__(128) void kg_fused_score_kernel(
    const float* __restrict__ ent, const float* __restrict__ W1,
    const float* __restrict__ W2, const float* __restrict__ b2,
    const float* __restrict__ ph, float* __restrict__ scores,
    int E, int B)
{
    __shared__ float As[MT * AS_STRIDE];        // 64x128 entity tile (padded)
    __shared__ float phs[8 * PH_STRIDE];        // ph tile (padded)
    __shared__ float w2s[HH];
    __shared__ float pts[16 * PT_STRIDE];       // one 16-entity pt subtile

    const int t  = threadIdx.x;
    const int e0 = blockIdx.x * MT;
    const int Bc = (B < 8) ? B : 8;

    // ---- stage A tile: 64 rows x 128 f32, coalesced float4 loads ----
    for (int i = t; i < MT * (DD / 4); i += 128) {
        const int row = i >> 5;                 // 32 float4s per row
        const int c4  = i & 31;
        int gr = e0 + row; if (gr >= E) gr = E - 1;   // clamp (dup rows harmless)
        const float4 v = ((const float4*)(ent + (size_t)gr * DD))[c4];
        float* dst = &As[row * AS_STRIDE + c4 * 4];
        dst[0] = v.x; dst[1] = v.y; dst[2] = v.z; dst[3] = v.w;
    }
    // ---- stage ph (padded) and W2 ----
    for (int i = t; i < Bc * HH; i += 128) {
        const int b = i >> 8, h = i & 255;
        phs[b * PH_STRIDE + h] = ph[(size_t)b * HH + h];
    }
    for (int i = t; i < HH; i += 128) w2s[i] = W2[i];
    __syncthreads();

    // ---- WMMA GEMM: pt(64x256) = A(64x128) * W1t(128x256) ----
    const int lane  = t & 31;
    const int wv    = t >> 5;
    const int mn    = lane & 15;                // A: row M;  B: col N
    const int khalf = (lane >> 4) * 2;          // lane half selects K pair
    const int n0    = wv * 64;                  // wave's first N column
    const float* Wt = W1 + (size_t)(2 * DD) * HH;   // W1t = W1[2D:, :]

    v8f acc[4][4];                              // [m-subtile][n-subtile]
    #pragma unroll
    for (int ms = 0; ms < 4; ++ms)
        #pragma unroll
        for (int nj = 0; nj < 4; ++nj)
            acc[ms][nj] = (v8f){};

    #pragma unroll 2
    for (int k0 = 0; k0 < DD; k0 += 4) {
        v2f a[4];
        #pragma unroll
        for (int ms = 0; ms < 4; ++ms) {
            const float* ap = &As[(ms * 16 + mn) * AS_STRIDE + k0 + khalf];
            a[ms].x = ap[0];  a[ms].y = ap[1];  // A[m][k], A[m][k+1]
        }
        const float* bp = Wt + (size_t)(k0 + khalf) * HH + n0 + mn;
        v2f bf[4];
        #pragma unroll
        for (int nj = 0; nj < 4; ++nj) {        // B[k][n], B[k+1][n]
            bf[nj].x = bp[nj * 16];
            bf[nj].y = bp[nj * 16 + HH];
        }
        #pragma unroll
        for (int ms = 0; ms < 4; ++ms)
            #pragma unroll
            for (int nj = 0; nj < 4; ++nj)
                acc[ms][nj] = __builtin_amdgcn_wmma_f32_16x16x4_f32(
                    false, a[ms], false, bf[nj], (short)0, acc[ms][nj],
                    false, false);
    }

    // ---- per M-subtile: stage pt to LDS, then score 16 entities x 8 batches
    const int mbase = (lane >> 4) * 8;          // D frag: vgpr i -> M = mbase+i
    const int el = t >> 3;                      // scoring: entity in subtile
    const int b  = t & 7;                       // scoring: batch
    #pragma unroll
    for (int ms = 0; ms < 4; ++ms) {
        __syncthreads();                        // pts free (prev scoring done)
        #pragma unroll
        for (int i = 0; i < 8; ++i) {
            float* pr = &pts[(mbase + i) * PT_STRIDE + n0 + mn];
            pr[0]  = acc[ms][0][i];
            pr[16] = acc[ms][1][i];
            pr[32] = acc[ms][2][i];
            pr[48] = acc[ms][3][i];
        }
        __syncthreads();                        // pts subtile complete
        const int e = e0 + ms * 16 + el;
        if (b < Bc && e < E) {
            float s = b2[0];
            const float* pr = &pts[el * PT_STRIDE];
            const float* hr = &phs[b * PH_STRIDE];
            #pragma unroll 8
            for (int h = 0; h < HH; ++h) {
                float v = hr[h] + pr[h];
                v = v > 0.0f ? v : 0.0f;
                s = fmaf(v, w2s[h], s);
            }
            scores[(size_t)b * E + e] = s;
        }
    }
}

// ---------------------------------------------------------------------------
// Kernel 3: per-batch top-k (k rounds of block argmax; tie -> lower index,
// matching lax.top_k).  Mutates the score scratch (rewritten every launch).
// ---------------------------------------------------------------------------
__global__ __launch_bounds__(256) void kg_topk_kernel(
    float* __restrict__ scores, int E, int k,
    int* __restrict__ out_idx, float* __restrict__ out_val)
{
    __shared__ float sv[256];
    __shared__ int   si[256];
    const int b = blockIdx.x;
    const int t = threadIdx.x;
    float* row = scores + (size_t)b * E;

    for (int r = 0; r < k; ++r) {
        float bs = -FLT_MAX;
        int   bi = 0x7fffffff;
        for (int i = t; i < E; i += 256) {
            const float s = row[i];
            if (s > bs || (s == bs && i < bi)) { bs = s; bi = i; }
        }
        sv[t] = bs; si[t] = bi;
        __syncthreads();
        for (int off = 128; off > 0; off >>= 1) {
            if (t < off) {
                const float os = sv[t + off];
                const int   oi = si[t + off];
                if (os > sv[t] || (os == sv[t] && oi < si[t])) { sv[t] = os; si[t] = oi; }
            }
            __syncthreads();
        }
        if (t == 0) {
            out_idx[b * k + r] = si[0];
            out_val[b * k + r] = sv[0];
            row[si[0]] = -FLT_MAX;              // exclude from later rounds
        }
        __syncthreads();
    }
}

// ---------------------------------------------------------------------------
// Launch.  Inputs: head, relation, k, ent_emb, rel_emb, W1, b1, W2, b2.
// Output: top_indices (B*k int32) then top_scores (B*k f32), concatenated.
// ---------------------------------------------------------------------------
extern "C" void kernel_launch(void* const* d_in, const int* in_sizes, int n_in,
                              void* d_out, int out_size, void* d_ws, size_t ws_size,
                              hipStream_t stream) {
    const int*   head     = (const int*)  d_in[0];
    const int*   relation = (const int*)  d_in[1];
    // d_in[2] is k on device; recover k on host from out_size instead.
    const float* ent      = (const float*)d_in[3];
    const float* rel      = (const float*)d_in[4];
    const float* W1       = (const float*)d_in[5];
    const float* b1       = (const float*)d_in[6];
    const float* W2       = (const float*)d_in[7];
    const float* b2       = (const float*)d_in[8];
    (void)n_in; (void)ws_size;

    const int B = in_sizes[0];
    const int E = in_sizes[3] / DD;
    const int k = out_size / (2 * B);           // out = B*k idx + B*k scores

    // workspace: ph (B*256 f32, 256B-aligned) then scores (B*E f32)
    float* ph_ws = (float*)d_ws;
    size_t ph_bytes = ((size_t)B * HH * sizeof(float) + 255) & ~(size_t)255;
    float* scores_ws = (float*)((char*)d_ws + ph_bytes);

    int*   out_idx = (int*)d_out;
    float* out_val = (float*)d_out + (size_t)B * k;

    kg_ph_kernel<<<B, 256, 0, stream>>>(head, relation, ent, rel, W1, b1, ph_ws);

    const int ntiles = (E + MT - 1) / MT;
    kg_fused_score_kernel<<<ntiles, 128, 0, stream>>>(ent, W1, W2, b2, ph_ws,
                                                      scores_ws, E, B);

    kg_topk_kernel<<<B, 256, 0, stream>>>(scores_ws, E, k, out_idx, out_val);
}